// SDG_63891933495864
// MI455X (gfx1250) — compile-verified
//
#include <hip/hip_runtime.h>
#include <stdint.h>

#define B_    32
#define N_    100
#define FIN   10
#define D_    64
#define NT    (B_*N_)
#define MPAD  112      // 100 padded up to 7*16 for WMMA tiles
#define ASTR  112      // att/sim row stride (floats)
#define ADJSTR 104
#define ALPHA_ 0.2f
#define EPS_  1e-8f
#define NEG_BIG -9.0e15f

typedef __attribute__((ext_vector_type(2))) float v2f;
typedef __attribute__((ext_vector_type(8))) float v8f;

// ---------------- JAX threefry2x32 (20 rounds), host+device ----------------
__host__ __device__ inline uint32_t rotl32(uint32_t x, int r) { return (x << r) | (x >> (32 - r)); }

__host__ __device__ inline void tf2x32(uint32_t k0, uint32_t k1, uint32_t x0, uint32_t x1,
                                       uint32_t* o0, uint32_t* o1) {
  uint32_t ks0 = k0, ks1 = k1, ks2 = k0 ^ k1 ^ 0x1BD11BDAu;
  x0 += ks0; x1 += ks1;
#define TFG(r0,r1,r2,r3,ka,kb,inc) \
  x0 += x1; x1 = rotl32(x1, r0); x1 ^= x0; \
  x0 += x1; x1 = rotl32(x1, r1); x1 ^= x0; \
  x0 += x1; x1 = rotl32(x1, r2); x1 ^= x0; \
  x0 += x1; x1 = rotl32(x1, r3); x1 ^= x0; \
  x0 += (ka); x1 += (kb) + (inc);
  TFG(13,15,26, 6, ks1, ks2, 1u)
  TFG(17,29,16,24, ks2, ks0, 2u)
  TFG(13,15,26, 6, ks0, ks1, 3u)
  TFG(17,29,16,24, ks1, ks2, 4u)
  TFG(13,15,26, 6, ks2, ks0, 5u)
#undef TFG
  *o0 = x0; *o1 = x1;
}

__device__ inline float uniform_from_bits(uint32_t bits) {
  // jax.random.uniform: bitcast((bits>>9)|0x3f800000) - 1, scaled to [1e-20, 1)
  float f = __uint_as_float((bits >> 9) | 0x3f800000u) - 1.0f;
  f = f + 1e-20f;
  return fmaxf(f, 1e-20f);
}

__device__ inline float gumbel_at(uint32_t k0, uint32_t k1, uint32_t f, uint32_t half) {
  // JAX threefry counter layout: pairs are (i, i + n/2)
  uint32_t a, b;
  if (f < half) { tf2x32(k0, k1, f, f + half, &a, &b); return -logf(-logf(uniform_from_bits(a))); }
  tf2x32(k0, k1, f - half, f, &a, &b);
  return -logf(-logf(uniform_from_bits(b)));
}

// ---------------- Kernel 0: static adjacency (shared by all samples) + zero BN sums ----
__global__ __launch_bounds__(256) void k_static(const float* __restrict__ node_emb,
                                                unsigned char* __restrict__ adjs,
                                                float* __restrict__ sums, float* __restrict__ sumsq,
                                                uint32_t k1a, uint32_t k1b) {
  __shared__ float inv[N_];
  int tid = threadIdx.x;
  if (tid < 2 * D_) { if (tid < D_) sums[tid] = 0.f; else sumsq[tid - D_] = 0.f; }
  if (tid < N_) {
    float s = 0.f;
    for (int d = 0; d < D_; ++d) { float v = node_emb[tid * D_ + d]; s += v * v; }
    inv[tid] = 1.0f / (sqrtf(s) + EPS_);
  }
  __syncthreads();
  for (int idx = tid; idx < N_ * N_; idx += 256) {
    int i = idx / N_, j = idx % N_;
    float dot = 0.f;
    for (int d = 0; d < D_; ++d) dot += node_emb[i * D_ + d] * node_emb[j * D_ + d];
    float s = dot * inv[i] * inv[j];                    // cosine sim
    uint32_t f0 = 2u * (uint32_t)idx;                   // flat idx into (100,100,2); n=20000
    float g0 = gumbel_at(k1a, k1b, f0, 10000u);
    float g1 = gumbel_at(k1a, k1b, f0 + 1u, 10000u);
    int on = (s + g0 >= 1.0f - s + g1) || (i == j);     // argmax==0  OR  +eye
    adjs[idx] = (unsigned char)on;
  }
}

// ---------------- Kernel 1: fused per-sample pipeline (one workgroup per sample) -------
__global__ __launch_bounds__(256) void k_gat(
    const float* __restrict__ data, const float* __restrict__ node_emb,
    const float* __restrict__ linW, const float* __restrict__ linb,
    const float* __restrict__ gsW, const float* __restrict__ gsb, const float* __restrict__ gsa,
    const float* __restrict__ gdW, const float* __restrict__ gdb, const float* __restrict__ gda,
    const unsigned char* __restrict__ adjs,
    float* __restrict__ outpre, float* __restrict__ sums, float* __restrict__ sumsq,
    uint32_t k2a, uint32_t k2b) {
  __shared__ float sh_x[N_ * FIN];             //  4.0 KB
  __shared__ float sh_h[MPAD * D_];            // 28.7 KB (n, then h_s, then h_d)
  __shared__ float sh_att[MPAD * ASTR];        // 50.2 KB (sim, then e/att per GAT)
  __shared__ float sh_out[N_ * D_];            // 25.6 KB
  __shared__ unsigned char sh_adjd[N_ * ADJSTR]; // 10.4 KB
  __shared__ float sh_left[N_], sh_right[N_], sh_rn[N_];

  const int b = blockIdx.x;
  const int tid = threadIdx.x;
  const int lane = tid & 31;
  const int wave = tid >> 5;      // 8 waves of 32 (wave32)
  const int ml = lane & 15;
  const int hi = lane >> 4;       // half-wave selector for f32 WMMA K striping

  for (int e = tid; e < N_ * FIN; e += 256) sh_x[e] = data[b * N_ * FIN + e];
  __syncthreads();

  // emb_d = x @ lin_W + lin_b  (pad rows 100..111 = 0)
  for (int e = tid; e < MPAD * D_; e += 256) {
    int r = e >> 6, d = e & 63;
    float acc = 0.f;
    if (r < N_) {
      acc = linb[d];
#pragma unroll
      for (int f = 0; f < FIN; ++f) acc += sh_x[r * FIN + f] * linW[f * D_ + d];
    }
    sh_h[e] = acc;
  }
  __syncthreads();
  if (tid < N_) {
    float s = 0.f;
    for (int d = 0; d < D_; ++d) { float v = sh_h[tid * D_ + d]; s += v * v; }
    sh_rn[tid] = 1.0f / (sqrtf(s) + EPS_);
  }
  __syncthreads();
  for (int e = tid; e < N_ * D_; e += 256) sh_h[e] *= sh_rn[e >> 6];
  __syncthreads();

  // sim_d = n @ n^T via V_WMMA_F32_16X16X4_F32 : 7x7 tiles of 16x16, K=64
  for (int t = wave; t < 49; t += 8) {
    int mt = t / 7, nt = t % 7;
    v8f acc = {0.f, 0.f, 0.f, 0.f, 0.f, 0.f, 0.f, 0.f};
#pragma unroll 4
    for (int kk = 0; kk < D_; kk += 4) {
      int k0i = kk + 2 * hi;
      v2f a, bv;
      a[0]  = sh_h[(mt * 16 + ml) * D_ + k0i + 0];
      a[1]  = sh_h[(mt * 16 + ml) * D_ + k0i + 1];
      bv[0] = sh_h[(nt * 16 + ml) * D_ + k0i + 0];   // B[k][j] = n[j][k]
      bv[1] = sh_h[(nt * 16 + ml) * D_ + k0i + 1];
      acc = __builtin_amdgcn_wmma_f32_16x16x4_f32(false, a, false, bv, (short)0, acc, false, false);
    }
#pragma unroll
    for (int r = 0; r < 8; ++r)
      sh_att[(mt * 16 + r + 8 * hi) * ASTR + nt * 16 + ml] = acc[r];
  }
  __syncthreads();

  // dynamic adjacency: gumbel-hard on sim (flat index into (3200,3200,2); n = 20480000)
  for (int idx = tid; idx < N_ * N_; idx += 256) {
    int i = idx / N_, j = idx % N_;
    float s = sh_att[i * ASTR + j];
    uint32_t p = (uint32_t)(b * N_ + i) * (uint32_t)NT + (uint32_t)(b * N_ + j);
    uint32_t f0 = 2u * p;
    float g0 = gumbel_at(k2a, k2b, f0, 10240000u);
    float g1 = gumbel_at(k2a, k2b, f0 + 1u, 10240000u);
    sh_adjd[i * ADJSTR + j] = (unsigned char)(s + g0 >= 1.0f - s + g1);
  }
  __syncthreads();

  // two GAT layers: g=0 static graph, g=1 dynamic graph
  for (int g = 0; g < 2; ++g) {
    const float* W  = g ? gdW : gsW;
    const float* bb = g ? gdb : gsb;
    const float* av = g ? gda : gsa;

    for (int e = tid; e < N_ * D_; e += 256) {       // h = x@W + b
      int r = e >> 6, d = e & 63;
      float acc = bb[d];
#pragma unroll
      for (int f = 0; f < FIN; ++f) acc += sh_x[r * FIN + f] * W[f * D_ + d];
      sh_h[e] = acc;
    }
    __syncthreads();
    if (tid < N_) {                                   // k = h + emb; left/right scores
      float al = 0.f, ar = 0.f;
      for (int d = 0; d < D_; ++d) {
        float kv = sh_h[tid * D_ + d] + node_emb[tid * D_ + d];
        al += kv * av[d];
        ar += kv * av[D_ + d];
      }
      sh_left[tid] = al; sh_right[tid] = ar;
    }
    __syncthreads();
    for (int e = tid; e < (MPAD - N_) * ASTR; e += 256) sh_att[N_ * ASTR + e] = 0.f; // pad rows
    if (tid < N_) {                                   // masked leaky-relu + softmax per row
      int r = tid;
      float lft = sh_left[r];
      float rmax = -3.0e38f;
      for (int j = 0; j < N_; ++j) {
        int on = g ? (int)sh_adjd[r * ADJSTR + j] : (int)adjs[r * N_ + j];
        float v = lft + sh_right[j];
        v = v > 0.f ? v : ALPHA_ * v;
        v = on ? v : NEG_BIG;
        rmax = fmaxf(rmax, v);
        sh_att[r * ASTR + j] = v;
      }
      float ssum = 0.f;
      for (int j = 0; j < N_; ++j) {
        float v = expf(sh_att[r * ASTR + j] - rmax);
        ssum += v;
        sh_att[r * ASTR + j] = v;
      }
      float inv = 1.0f / ssum;
      for (int j = 0; j < N_; ++j) sh_att[r * ASTR + j] *= inv;
    }
    __syncthreads();

    // hp = att @ h via WMMA (M tiles 7, N tiles 4, K=100 = 25 steps of 4), fused ELU
    for (int t = wave; t < 28; t += 8) {
      int mt = t >> 2, nt = t & 3;
      v8f acc = {0.f, 0.f, 0.f, 0.f, 0.f, 0.f, 0.f, 0.f};
      for (int kk = 0; kk < N_; kk += 4) {
        int k0i = kk + 2 * hi;
        v2f a, bv;
        a[0]  = sh_att[(mt * 16 + ml) * ASTR + k0i + 0];
        a[1]  = sh_att[(mt * 16 + ml) * ASTR + k0i + 1];
        bv[0] = sh_h[(k0i + 0) * D_ + nt * 16 + ml];
        bv[1] = sh_h[(k0i + 1) * D_ + nt * 16 + ml];
        acc = __builtin_amdgcn_wmma_f32_16x16x4_f32(false, a, false, bv, (short)0, acc, false, false);
      }
#pragma unroll
      for (int r = 0; r < 8; ++r) {
        int m = mt * 16 + r + 8 * hi;
        if (m < N_) {
          float v = acc[r];
          float e = v > 0.f ? v : (expf(v) - 1.0f);   // ELU
          int o = m * D_ + nt * 16 + ml;
          if (g == 0) sh_out[o] = e;
          else        sh_out[o] += 0.01f * e;
        }
      }
    }
    __syncthreads();
  }

  // out_pre = emb * node_emb ; per-channel BN partial sums (channel = tid&63 by construction)
  float ls = 0.f, lss = 0.f;
  for (int e = tid; e < N_ * D_; e += 256) {
    float v = sh_out[e] * node_emb[e];
    outpre[b * N_ * D_ + e] = v;
    ls += v; lss += v * v;
  }
  atomicAdd(&sums[tid & 63], ls);
  atomicAdd(&sumsq[tid & 63], lss);
}

// ---------------- Kernel 2: BatchNorm (batch stats) + ReLU + final projection ----------
__global__ __launch_bounds__(256) void k_out(
    const float* __restrict__ outpre, const float* __restrict__ sums, const float* __restrict__ sumsq,
    const float* __restrict__ gamma, const float* __restrict__ beta,
    const float* __restrict__ outW, const float* __restrict__ outb,
    float* __restrict__ out) {
  __shared__ float sm[D_], sc[D_];
  int tid = threadIdx.x;
  if (tid < D_) {
    float m = sums[tid] * (1.0f / NT);
    float var = sumsq[tid] * (1.0f / NT) - m * m;
    sm[tid] = m;
    sc[tid] = rsqrtf(var + 1e-5f) * gamma[tid];
  }
  __syncthreads();
  int i = blockIdx.x * 256 + tid;
  if (i < NT) {
    float acc = outb[0];
    for (int d = 0; d < D_; ++d) {
      float v = (outpre[i * D_ + d] - sm[d]) * sc[d] + beta[d];
      v = fmaxf(v, 0.f);
      acc += v * outW[d];
    }
    out[i] = acc;
  }
}

// ---------------- Launch ----------------
extern "C" void kernel_launch(void* const* d_in, const int* in_sizes, int n_in,
                              void* d_out, int out_size, void* d_ws, size_t ws_size,
                              hipStream_t stream) {
  const float* data     = (const float*)d_in[0];
  const float* node_emb = (const float*)d_in[1];
  const float* linW     = (const float*)d_in[2];
  const float* linb     = (const float*)d_in[3];
  const float* gsW      = (const float*)d_in[4];
  const float* gsb      = (const float*)d_in[5];
  const float* gsa      = (const float*)d_in[6];
  const float* gdW      = (const float*)d_in[7];
  const float* gdb      = (const float*)d_in[8];
  const float* gda      = (const float*)d_in[9];
  const float* gamma    = (const float*)d_in[10];
  const float* beta     = (const float*)d_in[11];
  const float* outW     = (const float*)d_in[12];
  const float* outb     = (const float*)d_in[13];
  (void)in_sizes; (void)n_in; (void)out_size; (void)ws_size;

  float* outpre = (float*)d_ws;                 // 3200*64 floats
  float* sums   = outpre + NT * D_;             // 64
  float* sumsq  = sums + D_;                    // 64
  unsigned char* adjs = (unsigned char*)(sumsq + D_); // 10000 bytes

  // JAX: key(42) -> split(2). counts [0,1,2,3] -> pairs (0,2),(1,3);
  // reshape(2,2): k1 = (p0.o0, p1.o0), k2 = (p0.o1, p1.o1)
  uint32_t a0, b0, a1, b1;
  tf2x32(0u, 42u, 0u, 2u, &a0, &b0);
  tf2x32(0u, 42u, 1u, 3u, &a1, &b1);
  uint32_t k1a = a0, k1b = a1, k2a = b0, k2b = b1;

  k_static<<<1, 256, 0, stream>>>(node_emb, adjs, sums, sumsq, k1a, k1b);
  k_gat<<<B_, 256, 0, stream>>>(data, node_emb, linW, linb, gsW, gsb, gsa,
                                gdW, gdb, gda, adjs, outpre, sums, sumsq, k2a, k2b);
  k_out<<<(NT + 255) / 256, 256, 0, stream>>>(outpre, sums, sumsq, gamma, beta, outW, outb,
                                              (float*)d_out);
}